// GCNBlock_19997367730291
// MI455X (gfx1250) — compile-verified
//
#include <hip/hip_runtime.h>

typedef __attribute__((ext_vector_type(2))) float v2f;
typedef __attribute__((ext_vector_type(8))) float v8f;

#define GD 128             // feature dimension D
#define ROWS_PER_BLOCK 128 // node rows per workgroup in the GEMM
#define GEMM_THREADS 256   // 8 waves
#define SX_PITCH 132       // x-strip pitch (words): 132 % 64 = 4 -> conflict-free A reads
#define SW_PITCH 288       // W pair-row pitch (words): 288 % 64 = 32 -> conflict-free B reads

// ---------------- kernel 1: deg = 1 (self-loop), agg = 0 -------------------
__global__ void k_init(float* __restrict__ deg, float* __restrict__ agg,
                       int N, int ND) {
    int gid = blockIdx.x * blockDim.x + threadIdx.x;
    if (gid < ND) agg[gid] = 0.0f;
    if (gid < N)  deg[gid] = 1.0f;
}

// ---------------- kernel 2: deg[dst] += ew ---------------------------------
__global__ void k_degree(const long long* __restrict__ ei,
                         const float* __restrict__ ew,
                         float* __restrict__ deg, int E) {
    int e = blockIdx.x * blockDim.x + threadIdx.x;
    if (e >= E) return;
    int dst = (int)ei[(size_t)E + e];   // edge_index[1][e]
    atomicAdd(&deg[dst], ew[e]);
}

// ---------------- kernel 3: deg -> rsqrt(deg); deg >= 1 always -------------
__global__ void k_dinv(float* __restrict__ deg, int N) {
    int i = blockIdx.x * blockDim.x + threadIdx.x;
    if (i < N) deg[i] = rsqrtf(deg[i]);
}

// ---------------- kernel 4: h = x @ W via V_WMMA_F32_16X16X4_F32 -----------
// W staged in LDS K-pair interleaved: each B fragment {W[kr][c], W[kr+1][c]}
// is one contiguous ds_load_b64 into an even-aligned VGPR pair (no shuffles).
__global__ __launch_bounds__(GEMM_THREADS)
void k_gemm(const float* __restrict__ x, const float* __restrict__ W,
            float* __restrict__ h, int N) {
    __shared__ float sX[ROWS_PER_BLOCK * SX_PITCH];  // 128*132*4 = 67584 B
    __shared__ float sWp[(GD / 2) * SW_PITCH];       // 64*288*4  = 73728 B (~138 KB total)

    const int tid  = threadIdx.x;
    const int wave = tid >> 5;
    const int lane = tid & 31;
    const int lh   = lane >> 4;   // half-wave select (0/1)
    const int lm   = lane & 15;
    const int rowBase = blockIdx.x * ROWS_PER_BLOCK;

    // stage W (coalesced float4 read, interleaved scatter into pair rows)
    for (int t = tid; t < GD * (GD / 4); t += GEMM_THREADS) {
        int row = t >> 5;
        int cf  = t & 31;
        float4 v = *(const float4*)(W + (size_t)row * GD + cf * 4);
        float* d = &sWp[(row >> 1) * SW_PITCH + (row & 1)];
        d[(cf * 4 + 0) * 2] = v.x;
        d[(cf * 4 + 1) * 2] = v.y;
        d[(cf * 4 + 2) * 2] = v.z;
        d[(cf * 4 + 3) * 2] = v.w;
    }
    // stage x strip (coalesced float4), zero-fill rows past N
    for (int t = tid; t < ROWS_PER_BLOCK * (GD / 4); t += GEMM_THREADS) {
        int row = t >> 5;           // GD/4 == 32 float4 per row
        int cf  = t & 31;
        float4 v = make_float4(0.f, 0.f, 0.f, 0.f);
        if (rowBase + row < N)
            v = *(const float4*)(x + (size_t)(rowBase + row) * GD + cf * 4);
        *(float4*)&sX[row * SX_PITCH + cf * 4] = v;
    }
    __syncthreads();

    const int r0 = wave * 16;       // this wave's 16-row tile within the strip
    v8f acc[8] = {};                // 8 column tiles of 16x16 f32

    for (int k = 0; k < GD; k += 4) {
        // A fragment 16x4: lanes 0-15 carry K=k..k+1, lanes 16-31 K=k+2..k+3
        v2f a = *(const v2f*)&sX[(r0 + lm) * SX_PITCH + k + 2 * lh];
        const float* bp = &sWp[((k >> 1) + lh) * SW_PITCH + lm * 2];
#pragma unroll
        for (int c = 0; c < 8; ++c) {
            // B fragment 4x16: contiguous pair {W[kr][col], W[kr+1][col]}
            v2f b = *(const v2f*)(bp + c * 32);
            acc[c] = __builtin_amdgcn_wmma_f32_16x16x4_f32(
                false, a, false, b, (short)0, acc[c], false, false);
        }
    }

    // epilogue: block-uniform fast path (only the last block is partial)
    float* hp = h + (size_t)(rowBase + r0 + lh * 8) * GD + lm;
    if (rowBase + ROWS_PER_BLOCK <= N) {
#pragma unroll
        for (int c = 0; c < 8; ++c)
#pragma unroll
            for (int i = 0; i < 8; ++i)     // VGPR i = row i (lo half) / i+8 (hi half)
                hp[(size_t)i * GD + c * 16] = acc[c][i];
    } else {
#pragma unroll
        for (int c = 0; c < 8; ++c)
#pragma unroll
            for (int i = 0; i < 8; ++i)
                if (rowBase + r0 + i + lh * 8 < N)
                    hp[(size_t)i * GD + c * 16] = acc[c][i];
    }
}

// ---------------- kernel 5: agg[dst] += h[src] * norm ----------------------
// one wave (32 lanes x float4) per edge; h is L2-resident (25.6 MB < 192 MB)
__global__ void k_scatter(const long long* __restrict__ ei,
                          const float* __restrict__ ew,
                          const float* __restrict__ dinv,
                          const float* __restrict__ h,
                          float* __restrict__ agg, int E) {
    long long gid = (long long)blockIdx.x * blockDim.x + threadIdx.x;
    int e    = (int)(gid >> 5);
    int lane = (int)(gid & 31);
    if (e >= E) return;
    int s = (int)ei[e];                 // edge_index[0][e]
    int t = (int)ei[(size_t)E + e];     // edge_index[1][e]
    float nrm = dinv[s] * ew[e] * dinv[t];
    float4 hv = *(const float4*)(h + (size_t)s * GD + lane * 4);
    float* ap = agg + (size_t)t * GD + lane * 4;
    atomicAdd(ap + 0, hv.x * nrm);
    atomicAdd(ap + 1, hv.y * nrm);
    atomicAdd(ap + 2, hv.z * nrm);
    atomicAdd(ap + 3, hv.w * nrm);
}

// ---------------- kernel 6: out = x + relu(agg + h*dinv^2 + b) -------------
// (self-loop contribution h[i]*dinv[i]^2 folded in here, no atomics needed)
__global__ void k_final(const float* __restrict__ x, const float* __restrict__ h,
                        const float* __restrict__ agg, const float* __restrict__ dinv,
                        const float* __restrict__ b, float* __restrict__ out, int ND) {
    int gid = blockIdx.x * blockDim.x + threadIdx.x;
    if (gid >= ND) return;
    int i = gid >> 7;       // node
    int d = gid & 127;      // feature
    float dv = dinv[i];
    float v = agg[gid] + h[gid] * (dv * dv) + b[d];
    out[gid] = x[gid] + fmaxf(v, 0.0f);
}

extern "C" void kernel_launch(void* const* d_in, const int* in_sizes, int n_in,
                              void* d_out, int out_size, void* d_ws, size_t ws_size,
                              hipStream_t stream) {
    const float*     x  = (const float*)d_in[0];
    const long long* ei = (const long long*)d_in[1];   // int64 per reference
    const float*     ew = (const float*)d_in[2];
    const float*     W  = (const float*)d_in[3];
    const float*     b  = (const float*)d_in[4];
    float*           out = (float*)d_out;

    const int ND = in_sizes[0];     // N * 128
    const int N  = ND / GD;
    const int E  = in_sizes[2];

    // workspace: [deg/dinv: N floats][h: N*D floats][agg: N*D floats] ~51.5 MB
    char*  ws = (char*)d_ws;
    size_t degBytes = (((size_t)N * sizeof(float)) + 255) & ~(size_t)255;
    float* deg = (float*)ws;
    float* h   = (float*)(ws + degBytes);
    float* agg = h + (size_t)N * GD;

    k_init  <<<(ND + 255) / 256, 256, 0, stream>>>(deg, agg, N, ND);
    k_degree<<<(E + 255) / 256, 256, 0, stream>>>(ei, ew, deg, E);
    k_dinv  <<<(N + 255) / 256, 256, 0, stream>>>(deg, N);
    k_gemm  <<<(N + ROWS_PER_BLOCK - 1) / ROWS_PER_BLOCK, GEMM_THREADS, 0, stream>>>(x, W, h, N);
    long long scatterThreads = (long long)E * 32;
    k_scatter<<<(int)((scatterThreads + 255) / 256), 256, 0, stream>>>(ei, ew, deg, h, agg, E);
    k_final <<<(ND + 255) / 256, 256, 0, stream>>>(x, h, agg, deg, b, out, ND);
}